// RBF_75299366633836
// MI455X (gfx1250) — compile-verified
//
#include <hip/hip_runtime.h>

typedef float v2f __attribute__((ext_vector_type(2)));
typedef float v8f __attribute__((ext_vector_type(8)));
typedef unsigned int v4u __attribute__((ext_vector_type(4)));
typedef int v4i __attribute__((ext_vector_type(4)));
typedef int v8i __attribute__((ext_vector_type(8)));

#define GAMMA 0.5f

// LDS layout (dynamic shared, no static __shared__ -> base offset 0):
//   xLds   : 128 rows x 68 floats (stride 68 = 64 + 4-dword TDM pad)   34,816 B
//   muPair : 32 rows  x 272 float2 (row k2 holds {mu[2k2][u], mu[2k2+1][u]})
//            544 dwords/row, 544%64==32 -> half-waves on disjoint banks 69,632 B
//   muSq   : 256 floats, xSq : 128 floats                               1,536 B
#define X_STRIDE   68
#define MU_STRIDE2 272
#define X_FLOATS   (128 * X_STRIDE)           // 8704
#define MU_FLOATS  (32 * MU_STRIDE2 * 2)      // 17408

__global__ __launch_bounds__(256) void rbf_wmma_kernel(
    const float* __restrict__ x,    // (65536, 64) row-major
    const float* __restrict__ mu,   // (64, 256)   row-major (d-major)
    float* __restrict__ out)        // (65536, 256)
{
    extern __shared__ float smem[];
    float*  xLds   = smem;                         // offset 0 in LDS
    float2* muPair = (float2*)(smem + X_FLOATS);
    float*  muSq   = smem + X_FLOATS + MU_FLOATS;  // 256
    float*  xSq    = muSq + 256;                   // 128

    const int tid    = threadIdx.x;
    const int lane   = tid & 31;
    const int wave   = tid >> 5;
    const int laneLo = lane & 15;
    const int hi     = lane >> 4;              // half-wave select
    const int blockRow = blockIdx.x * 128;

#if __has_builtin(__builtin_amdgcn_tensor_load_to_lds)
    // ---- TDM: DMA this block's 128x64 x-tile into LDS with stride-68 padding ----
    // Issued by wave 0 only (uniform branch; other waves jump over the tensor op).
    if (tid < 32) {
        const unsigned long long ga = (unsigned long long)(x + (size_t)blockRow * 64);
        // D# group 0: count=1 | lds_addr | global_addr[56:0] | type=2
        v4u g0 = { 1u,                                    // count = 1 valid descriptor
                   0u,                                    // lds_addr = 0 (xLds at LDS base)
                   (unsigned int)(ga & 0xFFFFFFFFu),
                   (unsigned int)((ga >> 32) & 0x01FFFFFFu) | 0x80000000u };
        // D# group 1: data_size=4B(2), pad_enable, pad_interval=64 dw(5), pad_amount=4 dw(3)
        v8i g1 = { (int)((2u << 16) | (1u << 20) | (5u << 22) | (3u << 25)),
                   (int)(64u << 16),    // tensor_dim0 = 64 (bits 79:48, low 16 here)
                   (int)(128u << 16),   // tensor_dim0 hi=0 | tensor_dim1 = 128
                   (int)(64u << 16),    // tensor_dim1 hi=0 | tile_dim0 = 64
                   (int)(128u),         // tile_dim1 = 128 | tile_dim2 = 0
                   (int)(64u),          // tensor_dim0_stride = 64
                   0, 0 };
        v4i gz4 = { 0, 0, 0, 0 };       // groups 2-3 unused (2-D tensor)
        v8i gz8 = { 0, 0, 0, 0, 0, 0, 0, 0 };
        // 6-arg toolchain variant: (g0, g1, g2, g3, g4, cpol)
        __builtin_amdgcn_tensor_load_to_lds(g0, g1, gz4, gz4, gz8, 0);
    }
#else
    // Fallback: manual coalesced staging of the x tile into stride-68 LDS.
    {
        const float4* x4b = reinterpret_cast<const float4*>(x + (size_t)blockRow * 64);
        for (int i = tid; i < 128 * 16; i += 256) {
            float4 v = x4b[i];
            int r  = i >> 4;
            int c4 = (i & 15) << 2;
            *reinterpret_cast<float4*>(&xLds[r * X_STRIDE + c4]) = v;
        }
    }
#endif

    // ---- stage mu into pair-interleaved LDS: muPair[k2][u] = {mu[2k2][u], mu[2k2+1][u]} ----
    __builtin_prefetch(mu + tid * 4, 0, 0);
    const float4* mu4 = reinterpret_cast<const float4*>(mu);
    for (int i = tid; i < 32 * 64; i += 256) {      // 2048 float4-pair groups
        int k2 = i >> 6;
        int cg = i & 63;
        float4 a = mu4[(2 * k2 + 0) * 64 + cg];
        float4 b = mu4[(2 * k2 + 1) * 64 + cg];
        float4 lo = { a.x, b.x, a.y, b.y };
        float4 hi4 = { a.z, b.z, a.w, b.w };
        float2* dst = &muPair[k2 * MU_STRIDE2 + (cg << 2)];
        *reinterpret_cast<float4*>(dst)     = lo;
        *reinterpret_cast<float4*>(dst + 2) = hi4;
    }

    // ---- ||mu_u||^2 per column (mu is 64KB, L2-resident across all 512 blocks) ----
    {
        float s = 0.f;
        for (int d = 0; d < 64; ++d) {
            float m = mu[d * 256 + tid];
            s += m * m;
        }
        muSq[tid] = s;
    }

    // ---- ||x_b||^2 for this block's 128 rows (coalesced; hits L2 behind the TDM) ----
    if (tid < 128) {
        const float4* x4 = reinterpret_cast<const float4*>(x + (size_t)(blockRow + tid) * 64);
        float s = 0.f;
#pragma unroll
        for (int i = 0; i < 16; ++i) {
            float4 v = x4[i];
            s += v.x * v.x + v.y * v.y + v.z * v.z + v.w * v.w;
        }
        xSq[tid] = s;
    }

#if __has_builtin(__builtin_amdgcn_tensor_load_to_lds)
    if (tid < 32) {
        __builtin_amdgcn_s_wait_tensorcnt(0);       // wave 0 owns the TENSORcnt
    }
#endif
    __syncthreads();                                // publishes TDM + staged LDS to all waves

    // ---- A fragments from LDS: 16x4 fp32 A layout per V_WMMA_F32_16X16X4_F32 ----
    //   lanes 0-15 : {K+0, K+1} (M=lane)   lanes 16-31: {K+2, K+3} (M=lane-16)
    // -> single ds_load_b64 per fragment, conflict-free at stride 68.
    const int rowLocal = wave * 16 + laneLo;
    const float* xfrag = &xLds[rowLocal * X_STRIDE + 2 * hi];
    v2f aFrag[16];
#pragma unroll
    for (int k = 0; k < 16; ++k) {
        aFrag[k] = *reinterpret_cast<const v2f*>(xfrag + 4 * k);
    }

    // ---- 16 N-tiles of 16 cols; 2 tiles in flight to break the WMMA D->C chain ----
    for (int n = 0; n < 16; n += 2) {
        const int col0 = n * 16 + laneLo;
        const int col1 = col0 + 16;

        // Software pipeline: issue ALL 32 B-fragment b64 loads for this n-pair up
        // front, then run the 32-WMMA chain. The scheduler can use progressive
        // s_wait_dscnt N waits instead of drain-to-zero before each WMMA.
        v2f bf0[16], bf1[16];
#pragma unroll
        for (int k = 0; k < 16; ++k) {
            // B fragment rows {4k+2hi, 4k+2hi+1} = pair row (2k + hi)
            const float2* bp = &muPair[(2 * k + hi) * MU_STRIDE2];
            bf0[k] = *reinterpret_cast<const v2f*>(bp + col0);
            bf1[k] = *reinterpret_cast<const v2f*>(bp + col1);
        }

        v8f acc0 = {0.f, 0.f, 0.f, 0.f, 0.f, 0.f, 0.f, 0.f};
        v8f acc1 = {0.f, 0.f, 0.f, 0.f, 0.f, 0.f, 0.f, 0.f};
#pragma unroll
        for (int k = 0; k < 16; ++k) {
            acc0 = __builtin_amdgcn_wmma_f32_16x16x4_f32(false, aFrag[k], false, bf0[k],
                                                         (short)0, acc0, false, false);
            acc1 = __builtin_amdgcn_wmma_f32_16x16x4_f32(false, aFrag[k], false, bf1[k],
                                                         (short)0, acc1, false, false);
        }

        // ---- fused RBF epilogue; C/D layout: VGPR v -> M = v + 8*hi, N = laneLo ----
        const float ms0 = muSq[col0];
        const float ms1 = muSq[col1];
#pragma unroll
        for (int v = 0; v < 8; ++v) {
            const int r = wave * 16 + v + hi * 8;
            const float xs = xSq[r];
            float* o = out + (size_t)(blockRow + r) * 256;
            o[col0] = __expf(-GAMMA * (xs - 2.f * acc0[v] + ms0));
            o[col1] = __expf(-GAMMA * (xs - 2.f * acc1[v] + ms1));
        }
    }
}

extern "C" void kernel_launch(void* const* d_in, const int* in_sizes, int n_in,
                              void* d_out, int out_size, void* d_ws, size_t ws_size,
                              hipStream_t stream) {
    const float* x  = (const float*)d_in[0];   // (65536, 64)
    const float* mu = (const float*)d_in[1];   // (64, 256)
    float* out = (float*)d_out;                // (65536, 256)

    const int B = in_sizes[0] / 64;            // 65536 rows
    dim3 grid(B / 128);
    dim3 block(256);
    const size_t ldsBytes = (size_t)(X_FLOATS + MU_FLOATS + 256 + 128) * sizeof(float);
    rbf_wmma_kernel<<<grid, block, ldsBytes, stream>>>(x, mu, out);
}